// SwitchFeedForward_79096117723389
// MI455X (gfx1250) — compile-verified
//
#include <hip/hip_runtime.h>
#include <math.h>

// Problem constants (fixed by the reference)
#define S_ 2048
#define B_ 2
#define D_ 1024
#define E_ 8
#define F_ 4096
#define N_ (S_ * B_)                 // 4096 tokens
#define TILE_M 64
#define TILE_N 256
#define KSTEP 32
#define NPAD (N_ + E_ * TILE_M)      // 4608 (per-expert segments padded to TILE_M)
#define MAX_TILES (NPAD / TILE_M)    // 72

typedef float v2f __attribute__((ext_vector_type(2)));
typedef float v8f __attribute__((ext_vector_type(8)));
typedef int   v4i __attribute__((ext_vector_type(4)));

#define AS1 __attribute__((address_space(1)))
#define AS3 __attribute__((address_space(3)))

#if __has_builtin(__builtin_amdgcn_global_load_async_to_lds_b128)
#define HAVE_ASYNC 1
#else
#define HAVE_ASYNC 0
#endif

__device__ __forceinline__ void wait_async0() {
#if __has_builtin(__builtin_amdgcn_s_wait_asynccnt)
  __builtin_amdgcn_s_wait_asynccnt(0);
#else
  asm volatile("s_wait_asynccnt 0x0" ::: "memory");
#endif
}

__device__ __forceinline__ v8f wmma_f32(v2f a, v2f b, v8f c) {
  return __builtin_amdgcn_wmma_f32_16x16x4_f32(false, a, false, b, (short)0, c,
                                               false, false);
}

// ---------------------------------------------------------------- init
__global__ void init_kernel(int* counts_i, int* cursors) {
  int t = threadIdx.x;
  if (t < E_) { counts_i[t] = 0; cursors[t] = 0; }
}

// ---------------------------------------------------------------- router
__global__ __launch_bounds__(256)
void router_kernel(const float* __restrict__ x, const float* __restrict__ wsw,
                   const float* __restrict__ bsw, int* __restrict__ routes,
                   float* __restrict__ scaleArr, float* __restrict__ probbuf,
                   int* __restrict__ counts_i) {
  int wave = threadIdx.x >> 5;
  int lane = threadIdx.x & 31;
  int n = blockIdx.x * 8 + wave;
  if (n >= N_) return;
  const float* xr = x + (size_t)n * D_;
  float acc[E_];
#pragma unroll
  for (int e = 0; e < E_; ++e) acc[e] = 0.f;
  for (int d = lane; d < D_; d += 32) {
    float xv = xr[d];
    const float* wr = wsw + d * E_;
#pragma unroll
    for (int e = 0; e < E_; ++e) acc[e] += xv * wr[e];
  }
#pragma unroll
  for (int e = 0; e < E_; ++e)
    for (int off = 16; off > 0; off >>= 1)
      acc[e] += __shfl_down(acc[e], off, 32);
  if (lane == 0) {
    float logit[E_], p[E_];
    float m = -1e30f;
#pragma unroll
    for (int e = 0; e < E_; ++e) { logit[e] = acc[e] + bsw[e]; m = fmaxf(m, logit[e]); }
    float ssum = 0.f;
#pragma unroll
    for (int e = 0; e < E_; ++e) { p[e] = __expf(logit[e] - m); ssum += p[e]; }
    float inv = 1.f / ssum;
    float pmax = -1.f; int arg = 0;
#pragma unroll
    for (int e = 0; e < E_; ++e) {
      p[e] *= inv;
      probbuf[(size_t)n * E_ + e] = p[e];
      if (p[e] > pmax) { pmax = p[e]; arg = e; }   // strict > keeps first index
    }
    routes[n] = arg;
    scaleArr[n] = pmax;
    atomicAdd(&counts_i[arg], 1);
  }
}

// ---------------------------------------------------------------- compact
__global__ __launch_bounds__(256)
void compact_kernel(const int* __restrict__ routes, const float* __restrict__ probbuf,
                    const int* __restrict__ counts_i, int* __restrict__ cursors,
                    int* __restrict__ tokmap, int* __restrict__ tile_expert,
                    float* __restrict__ out_tail) {
  __shared__ int seg_s[E_ + 1];
  int tid = threadIdx.x;
  if (tid == 0) {
    int off = 0;
    for (int e = 0; e < E_; ++e) {
      seg_s[e] = off;
      int c = counts_i[e];
      off += ((c + TILE_M - 1) / TILE_M) * TILE_M;
    }
    seg_s[E_] = off;
    for (int t = 0; t < MAX_TILES; ++t) {
      int row = t * TILE_M;
      int et = -1;
      if (row < seg_s[E_]) {
        for (int e = 0; e < E_; ++e)
          if (row >= seg_s[e] && row < seg_s[e + 1]) { et = e; break; }
      }
      tile_expert[t] = et;
    }
    for (int e = 0; e < E_; ++e) out_tail[e] = (float)counts_i[e];  // counts (fp32)
    ((int*)out_tail)[2 * E_] = 0;                                    // n_dropped
  }
  if (tid < E_) {  // deterministic route_prob.sum(0)
    float s = 0.f;
    for (int n = 0; n < N_; ++n) s += probbuf[(size_t)n * E_ + tid];
    out_tail[E_ + tid] = s;
  }
  __syncthreads();
  for (int i = tid; i < NPAD; i += 256) tokmap[i] = -1;
  __syncthreads();
  for (int n = tid; n < N_; n += 256) {
    int e = routes[n];
    int pos = atomicAdd(&cursors[e], 1);
    tokmap[seg_s[e] + pos] = n;   // order nondeterministic; output values invariant
  }
}

// ---------------------------------------------------------------- GEMM
// Block: 256 thr = 8 waves (2 M x 4 N), block tile 64x256, wave tile 32x64,
// double-buffered LDS, async global->LDS for the weight tile.
template <int KTOT, int NCOL, bool GATHER, bool RELU, bool SCATTER>
__global__ __launch_bounds__(256)
void gemm_kernel(const float* __restrict__ Abase, const float* __restrict__ scaleArr,
                 const int* __restrict__ tokmap, const int* __restrict__ tile_expert,
                 const float* __restrict__ W, const float* __restrict__ bias,
                 float* __restrict__ Out) {
  int tileM = blockIdx.y;
  int expert = tile_expert[tileM];
  if (expert < 0) return;
  int n0 = blockIdx.x * TILE_N;

  __shared__ float As[2][TILE_M][KSTEP + 2];   // stride 34: conflict-free frag reads
  __shared__ float Bs[2][KSTEP][TILE_N];       // stride 256: conflict-free + 16B aligned
  __shared__ int   tok_s[TILE_M];
  __shared__ float scl_s[TILE_M];

  int tid = threadIdx.x;
  int lane = tid & 31;
  int wave = tid >> 5;
  int wm = wave >> 2;   // 0..1
  int wn = wave & 3;    // 0..3
  int am = lane & 15;

  if (tid < TILE_M) {
    int t = tokmap[tileM * TILE_M + tid];
    tok_s[tid] = t;
    scl_s[tid] = (t >= 0) ? scaleArr[t] : 0.f;
  }
  __syncthreads();

  const float* Wbase = W + (size_t)expert * KTOT * NCOL;

  // Per-thread staging assignments
  int ar = tid >> 2;              // A row (0..63), 8 consecutive K elems
  int akq = (tid & 3) * 8;
  int bk = tid >> 3;              // B k-row (0..31), 32 consecutive N elems
  int bnq = (tid & 7) * 32;

  const float* aptr;
  float ascale = 1.f;
  bool avalid = true;
  if (GATHER) {
    int t = tok_s[ar];
    avalid = (t >= 0);
    aptr = Abase + (size_t)(avalid ? t : 0) * KTOT + akq;
    ascale = scl_s[ar];
  } else {
    aptr = Abase + ((size_t)tileM * TILE_M + ar) * KTOT + akq;
  }

  v8f zero = {0.f, 0.f, 0.f, 0.f, 0.f, 0.f, 0.f, 0.f};
  v8f acc[2][4];
#pragma unroll
  for (int i = 0; i < 2; ++i)
#pragma unroll
    for (int j = 0; j < 4; ++j) acc[i][j] = zero;

  float4 au, av;

  auto loadA = [&](int kb) {
    if (!GATHER || avalid) {
      const float4* p = (const float4*)(aptr + kb);
      au = p[0];
      av = p[1];
    } else {
      au = make_float4(0.f, 0.f, 0.f, 0.f);
      av = make_float4(0.f, 0.f, 0.f, 0.f);
    }
  };
  auto storeA = [&](int b) {
    float s = GATHER ? ascale : 1.f;
    As[b][ar][akq + 0] = au.x * s; As[b][ar][akq + 1] = au.y * s;
    As[b][ar][akq + 2] = au.z * s; As[b][ar][akq + 3] = au.w * s;
    As[b][ar][akq + 4] = av.x * s; As[b][ar][akq + 5] = av.y * s;
    As[b][ar][akq + 6] = av.z * s; As[b][ar][akq + 7] = av.w * s;
  };
  auto loadB = [&](int kb, int b) {
    const float* wr = Wbase + (size_t)(kb + bk) * NCOL + n0 + bnq;
#if HAVE_ASYNC
#pragma unroll
    for (int j = 0; j < 8; ++j)
      __builtin_amdgcn_global_load_async_to_lds_b128(
          (AS1 v4i*)(wr + 4 * j), (AS3 v4i*)&Bs[b][bk][bnq + 4 * j], 0, 0);
#else
#pragma unroll
    for (int j = 0; j < 8; ++j)
      *(float4*)&Bs[b][bk][bnq + 4 * j] = *(const float4*)(wr + 4 * j);
#endif
  };
  auto compute = [&](int b) {
#pragma unroll
    for (int kk = 0; kk < KSTEP / 4; ++kk) {
      int k0 = kk * 4 + 2 * (lane >> 4);   // lanes 0-15: K=4kk; 16-31: K=4kk+2
      v2f a0, a1;
      a0.x = As[b][wm * 32 + am][k0];        a0.y = As[b][wm * 32 + am][k0 + 1];
      a1.x = As[b][wm * 32 + 16 + am][k0];   a1.y = As[b][wm * 32 + 16 + am][k0 + 1];
#pragma unroll
      for (int fn = 0; fn < 4; ++fn) {
        int bc = wn * 64 + fn * 16 + am;
        v2f bb;
        bb.x = Bs[b][k0][bc];
        bb.y = Bs[b][k0 + 1][bc];
        acc[0][fn] = wmma_f32(a0, bb, acc[0][fn]);
        acc[1][fn] = wmma_f32(a1, bb, acc[1][fn]);
      }
    }
  };

  constexpr int NST = KTOT / KSTEP;
  // Prologue: stage 0
  loadB(0, 0);
  loadA(0);
  storeA(0);
  wait_async0();
  __syncthreads();

  for (int i = 0; i < NST; ++i) {
    int b = i & 1;
    bool next = (i + 1 < NST);
    if (next) {
      loadB((i + 1) * KSTEP, b ^ 1);   // async into other buffer
      loadA((i + 1) * KSTEP);          // global loads into regs
    }
    compute(b);                        // WMMA on current buffer (overlaps loads)
    if (next) {
      storeA(b ^ 1);                   // regs -> LDS (other buffer)
      wait_async0();
    }
    __syncthreads();
  }

  const float* bp = bias + (size_t)expert * NCOL;
#pragma unroll
  for (int fm = 0; fm < 2; ++fm) {
#pragma unroll
    for (int fn = 0; fn < 4; ++fn) {
      int col = n0 + wn * 64 + fn * 16 + am;
      float bv = bp[col];
      int rbase = wm * 32 + fm * 16 + ((lane >> 4) << 3);
#pragma unroll
      for (int v = 0; v < 8; ++v) {
        int r = rbase + v;
        float val = acc[fm][fn][v] + bv;
        if (RELU) val = fmaxf(val, 0.f);
        if (SCATTER) {
          int t = tok_s[r];
          if (t >= 0) Out[(size_t)t * NCOL + col] = val;
        } else {
          Out[((size_t)tileM * TILE_M + r) * NCOL + col] = val;
        }
      }
    }
  }
}

// ---------------------------------------------------------------- launch
extern "C" void kernel_launch(void* const* d_in, const int* in_sizes, int n_in,
                              void* d_out, int out_size, void* d_ws, size_t ws_size,
                              hipStream_t stream) {
  const float* x   = (const float*)d_in[0];
  const float* wsw = (const float*)d_in[1];
  const float* bsw = (const float*)d_in[2];
  const float* w1  = (const float*)d_in[3];
  const float* b1  = (const float*)d_in[4];
  const float* w2  = (const float*)d_in[5];
  const float* b2  = (const float*)d_in[6];
  float* out = (float*)d_out;

  char* wsp = (char*)d_ws;
  auto alloc = [&](size_t bytes) -> char* {
    char* p = wsp;
    wsp += (bytes + 255) & ~(size_t)255;
    return p;
  };
  int*   routes      = (int*)  alloc((size_t)N_ * 4);
  float* scaleArr    = (float*)alloc((size_t)N_ * 4);
  float* probbuf     = (float*)alloc((size_t)N_ * E_ * 4);
  int*   tokmap      = (int*)  alloc((size_t)NPAD * 4);
  int*   counts_i    = (int*)  alloc(E_ * 4);
  int*   cursors     = (int*)  alloc(E_ * 4);
  int*   tile_expert = (int*)  alloc(MAX_TILES * 4);
  float* H           = (float*)alloc((size_t)NPAD * F_ * 4);

  init_kernel<<<1, 32, 0, stream>>>(counts_i, cursors);
  router_kernel<<<N_ / 8, 256, 0, stream>>>(x, wsw, bsw, routes, scaleArr,
                                            probbuf, counts_i);
  compact_kernel<<<1, 256, 0, stream>>>(routes, probbuf, counts_i, cursors,
                                        tokmap, tile_expert,
                                        out + (size_t)N_ * D_);
  // ffn1: H = relu( gathered(x)*scale @ w1[e] + b1[e] )
  gemm_kernel<D_, F_, true, true, false>
      <<<dim3(F_ / TILE_N, MAX_TILES), 256, 0, stream>>>(
          x, scaleArr, tokmap, tile_expert, w1, b1, H);
  // ffn2: out[tok] = H @ w2[e] + b2[e]
  gemm_kernel<F_, D_, false, false, true>
      <<<dim3(D_ / TILE_N, MAX_TILES), 256, 0, stream>>>(
          H, scaleArr, tokmap, tile_expert, w2, b2, out);
}